// REDFM_15676630630653
// MI455X (gfx1250) — compile-verified
//
#include <hip/hip_runtime.h>
#include <cstdint>

// Problem constants (from reference: B=4, K=8192, CG=1024, G=8)
#define NKP    32768          // B*K keypoints
#define CG_    1024           // channels per keypoint
#define WPB    4              // waves per block
#define THR    (WPB * 32)     // 128 threads
#define LANE_W 36             // padded words per lane in LDS (32 data + 4 pad -> kills bank conflicts)
#define REG_W  (32 * LANE_W)  // 1152 words per buffer per wave
#define BUF_B  (REG_W * 4)    // 4608 bytes per buffer

typedef __attribute__((ext_vector_type(4))) float f4;

// Roll each 8-float group left by S (compile-time), scale already applied; store 2x b128 per group.
template<int S>
__device__ __forceinline__ void store_rolled(f4* __restrict__ dst, const float* v) {
#pragma unroll
  for (int q = 0; q < 4; ++q) {
    const float* g = v + q * 8;
    f4 lo = { g[(0 + S) & 7], g[(1 + S) & 7], g[(2 + S) & 7], g[(3 + S) & 7] };
    f4 hi = { g[(4 + S) & 7], g[(5 + S) & 7], g[(6 + S) & 7], g[(7 + S) & 7] };
    __builtin_nontemporal_store(lo, dst + 2 * q + 0);
    __builtin_nontemporal_store(hi, dst + 2 * q + 1);
  }
}

__global__ __launch_bounds__(THR) void desc_roll_norm(const float* __restrict__ desc,
                                                      float* __restrict__ outd) {
  __shared__ float smem[WPB * 2 * REG_W];   // 36864 bytes: 2 buffers per wave

  const int lane = threadIdx.x & 31;
  const int w    = threadIdx.x >> 5;
  const int wave = (int)((blockIdx.x * THR + threadIdx.x) >> 5);
  const int nw   = (int)((gridDim.x * THR) >> 5);

  // Low 32 bits of the generic LDS address == wave-relative LDS byte offset (ISA 10.2)
  const uint32_t lds0 = (uint32_t)(uintptr_t)(&smem[0]) + (uint32_t)(w * 2 * BUF_B);

  // Issue one 4KB keypoint tile: 8 async b128 transfers, contiguous on the global side
  // (issue i covers bytes [i*512, i*512+512) of the keypoint), lane-blocked+padded in LDS.
  auto issue = [&](int kp, int b) {
    const float* g = desc + (size_t)kp * CG_;
    const uint32_t lb = lds0 + (uint32_t)(b * BUF_B);
#pragma unroll
    for (int i = 0; i < 8; ++i) {
      // global channels i*128 + lane*4 .. +3 belong to owner lane L = i*4 + lane/8
      uint32_t la = lb + 4u * (uint32_t)((i * 4 + (lane >> 3)) * LANE_W + (lane & 7) * 4);
      uint64_t ga = (uint64_t)(uintptr_t)(g + i * 128 + lane * 4);
      asm volatile("global_load_async_to_lds_b128 %0, %1, off"
                   :: "v"(la), "v"(ga) : "memory");
    }
  };

  int buf = 0;
  if (wave < NKP) issue(wave, 0);

  for (int kp = wave; kp < NKP; kp += nw) {
    const int nxt = kp + nw;
    if (nxt < NKP) {
      issue(nxt, buf ^ 1);                                  // prefetch next tile into other buffer
      asm volatile("s_wait_asynccnt 0x8" ::: "memory");     // current tile's 8 ops complete
    } else {
      asm volatile("s_wait_asynccnt 0x0" ::: "memory");
    }

    // Read back my 32 channels (channels lane*32 .. +31) — padded stride => conflict-free b128s
    float v[32];
    const f4* src = (const f4*)(&smem[(size_t)(w * 2 + buf) * REG_W + lane * LANE_W]);
#pragma unroll
    for (int i = 0; i < 8; ++i) {
      f4 t = src[i];
      v[i * 4 + 0] = t.x; v[i * 4 + 1] = t.y; v[i * 4 + 2] = t.z; v[i * 4 + 3] = t.w;
    }

    // argmax of channels 0..7 (first-match, strict >) — valid on lane 0, broadcast as SGPR
    int sl = 0; float bb = v[0];
#pragma unroll
    for (int j = 1; j < 8; ++j) if (v[j] > bb) { bb = v[j]; sl = j; }
    const int s = __builtin_amdgcn_readlane(sl, 0);

    // sum of squares (roll-invariant) + wave32 reduction
    float ss = 0.f;
#pragma unroll
    for (int j = 0; j < 32; ++j) ss = fmaf(v[j], v[j], ss);
#pragma unroll
    for (int o = 16; o > 0; o >>= 1) ss += __shfl_xor(ss, o, 32);

    const float inv = 1.0f / (sqrtf(ss) + 1e-6f);
#pragma unroll
    for (int j = 0; j < 32; ++j) v[j] *= inv;

    f4* dst = (f4*)(outd + (size_t)kp * CG_ + lane * 32);
    switch (s) {                      // s is uniform (SGPR) -> scalar branches, permute is free
      case 0: store_rolled<0>(dst, v); break;
      case 1: store_rolled<1>(dst, v); break;
      case 2: store_rolled<2>(dst, v); break;
      case 3: store_rolled<3>(dst, v); break;
      case 4: store_rolled<4>(dst, v); break;
      case 5: store_rolled<5>(dst, v); break;
      case 6: store_rolled<6>(dst, v); break;
      default: store_rolled<7>(dst, v); break;
    }
    buf ^= 1;
  }
}

__global__ void copy_kpts(const f4* __restrict__ a, f4* __restrict__ b, int n4) {
  int i = blockIdx.x * blockDim.x + threadIdx.x;
  if (i < n4) b[i] = a[i];
}

extern "C" void kernel_launch(void* const* d_in, const int* in_sizes, int n_in,
                              void* d_out, int out_size, void* d_ws, size_t ws_size,
                              hipStream_t stream) {
  (void)in_sizes; (void)n_in; (void)out_size; (void)d_ws; (void)ws_size;
  const float* kpts = (const float*)d_in[0];   // [B,K,2]  = 65536 floats
  const float* desc = (const float*)d_in[1];   // [B,K,CG] = 33554432 floats
  float* out = (float*)d_out;                  // kpts_update ++ desc_update

  const int n4 = (NKP * 2) / 4;                // 16384 float4s of kpts
  copy_kpts<<<(n4 + 255) / 256, 256, 0, stream>>>((const f4*)kpts, (f4*)out, n4);

  // 1024 blocks x 4 waves = 4096 waves; 8 keypoints per wave, no partial waves.
  desc_roll_norm<<<1024, THR, 0, stream>>>(desc, out + NKP * 2);
}